// GCN_69269232550026
// MI455X (gfx1250) — compile-verified
//
#include <hip/hip_runtime.h>

#define C_CH 64

typedef float v2f __attribute__((ext_vector_type(2)));
typedef float v8f __attribute__((ext_vector_type(8)));

__global__ void k_fill(float* __restrict__ p, float v, int n) {
  int i = blockIdx.x * blockDim.x + threadIdx.x;
  if (i < n) p[i] = v;
}

// degree accumulation (self-loop already seeded as 1.0 by k_fill)
__global__ void k_deg(const int* __restrict__ dst, float* __restrict__ deg, int E) {
  int e = blockIdx.x * blockDim.x + threadIdx.x;
  if (e < E) atomicAdd(&deg[dst[e]], 1.0f);
}

__global__ void k_rsqrt(float* __restrict__ d, int n) {
  int i = blockIdx.x * blockDim.x + threadIdx.x;
  if (i < n) d[i] = rsqrtf(d[i]);   // deg >= 1 always (self-loops)
}

// H[nrows,64] = X[nrows,K] @ W[K,64] using V_WMMA_F32_16X16X4_F32.
// One wave computes a 16x64 strip (4 col-tiles). Wave-uniform exit keeps EXEC full.
template<int K>
__global__ void k_gemm_wmma(const float* __restrict__ X, const float* __restrict__ W,
                            float* __restrict__ H, int nrows) {
  int wave = (blockIdx.x * blockDim.x + threadIdx.x) >> 5;
  int lane = threadIdx.x & 31;
  int row0 = wave << 4;
  if (row0 >= nrows) return;
  int m  = lane & 15;   // A row / B,D col
  int hi = lane >> 4;   // lane half
  v8f acc0 = {}, acc1 = {}, acc2 = {}, acc3 = {};
  const float* xrow = X + (size_t)(row0 + m) * K;
#pragma unroll 4
  for (int kb = 0; kb < K; kb += 4) {
    int k0 = kb + (hi << 1);
    v2f a = *(const v2f*)(xrow + k0);                 // A: K = k0, k0+1
    const float* wr0 = W + (size_t)k0 * C_CH + m;     // B VGPR0 row
    const float* wr1 = wr0 + C_CH;                    // B VGPR1 row
    v2f b0 = { wr0[0],  wr1[0]  };
    v2f b1 = { wr0[16], wr1[16] };
    v2f b2 = { wr0[32], wr1[32] };
    v2f b3 = { wr0[48], wr1[48] };
    acc0 = __builtin_amdgcn_wmma_f32_16x16x4_f32(false, a, false, b0, (short)0, acc0, false, false);
    acc1 = __builtin_amdgcn_wmma_f32_16x16x4_f32(false, a, false, b1, (short)0, acc1, false, false);
    acc2 = __builtin_amdgcn_wmma_f32_16x16x4_f32(false, a, false, b2, (short)0, acc2, false, false);
    acc3 = __builtin_amdgcn_wmma_f32_16x16x4_f32(false, a, false, b3, (short)0, acc3, false, false);
  }
  float* hb = H + (size_t)row0 * C_CH + m;
#pragma unroll
  for (int v = 0; v < 8; ++v) {
    float* r = hb + (size_t)(v + (hi << 3)) * C_CH;   // D: row = v + 8*hi
    r[0]  = acc0[v];
    r[16] = acc1[v];
    r[32] = acc2[v];
    r[48] = acc3[v];
  }
}

// agg[i,c] = h[i,c]*dis[i]^2 + b[c]   (self-loop message + bias)
__global__ void k_self(const float* __restrict__ h, const float* __restrict__ dis,
                       const float* __restrict__ b, float* __restrict__ agg, int n) {
  int t = blockIdx.x * blockDim.x + threadIdx.x;
  if (t >= n * C_CH) return;
  int i = t >> 6, c = t & 63;
  float di = dis[i];
  agg[t] = h[t] * di * di + b[c];
}

// 16 lanes per edge, float4 gather + 4x global_atomic_add_f32 scatter
__global__ void k_edge(const int* __restrict__ src, const int* __restrict__ dst,
                       const float* __restrict__ dis, const float* __restrict__ h,
                       float* __restrict__ agg, int E) {
  int t = blockIdx.x * blockDim.x + threadIdx.x;
  int e = t >> 4;
  if (e >= E) return;
  int c4 = (t & 15) << 2;
  int s = src[e], d = dst[e];
  float w = dis[s] * dis[d];
  const float4 hv = *(const float4*)(h + (size_t)s * C_CH + c4);
  float* o = agg + (size_t)d * C_CH + c4;
  atomicAdd(o + 0, hv.x * w);
  atomicAdd(o + 1, hv.y * w);
  atomicAdd(o + 2, hv.z * w);
  atomicAdd(o + 3, hv.w * w);
}

__global__ void k_relu(const float* __restrict__ a, float* __restrict__ o, int n) {
  int t = blockIdx.x * blockDim.x + threadIdx.x;
  if (t < n) o[t] = fmaxf(a[t], 0.0f);
}

// global_add_pool via atomics (batch sorted -> L2-resident accumulators)
__global__ void k_pool(const float* __restrict__ h, const int* __restrict__ batch,
                       float* __restrict__ p, int n) {
  int t = blockIdx.x * blockDim.x + threadIdx.x;
  if (t >= n * C_CH) return;
  int i = t >> 6, c = t & 63;
  atomicAdd(&p[batch[i] * C_CH + c], h[t]);
}

// z[g,k] = bl[k] + sum_c p[g,c]*Wl[c,k]   (64x64x10 -> scalar is fine)
__global__ void k_fc(const float* __restrict__ p, const float* __restrict__ Wl,
                     const float* __restrict__ bl, float* __restrict__ z,
                     int G, int CLS) {
  int t = blockIdx.x * blockDim.x + threadIdx.x;
  if (t >= G * CLS) return;
  int g = t / CLS, k = t % CLS;
  float acc = bl[k];
  for (int c = 0; c < C_CH; ++c) acc += p[g * C_CH + c] * Wl[c * CLS + k];
  z[t] = acc;
}

extern "C" void kernel_launch(void* const* d_in, const int* in_sizes, int n_in,
                              void* d_out, int out_size, void* d_ws, size_t ws_size,
                              hipStream_t stream) {
  (void)n_in; (void)ws_size;
  const int F_IN = 128, CLS = 10;
  const float* x   = (const float*)d_in[0];
  const int*   ei  = (const int*)d_in[1];
  const int*   bat = (const int*)d_in[2];
  const float* Wm[4] = {(const float*)d_in[3], (const float*)d_in[5],
                        (const float*)d_in[7], (const float*)d_in[9]};
  const float* bv[4] = {(const float*)d_in[4], (const float*)d_in[6],
                        (const float*)d_in[8], (const float*)d_in[10]};
  const float* Wl = (const float*)d_in[11];
  const float* bl = (const float*)d_in[12];

  int N = in_sizes[0] / F_IN;          // 100000
  int E = in_sizes[1] / 2;             // 1600000
  int G = (out_size - 4 * N * C_CH) / CLS;  // 64

  const int* src = ei;
  const int* dst = ei + E;

  float* ws  = (float*)d_ws;
  float* dis = ws;
  float* h   = dis + N;                       // N % 4 == 0 -> 16B aligned
  float* agg = h   + (size_t)N * C_CH;
  float* pp  = agg + (size_t)N * C_CH;

  float* out = (float*)d_out;
  size_t NC = (size_t)N * C_CH;
  float* outs[4] = { out, out + NC, out + 2 * NC, out + 3 * NC };
  float* zout = out + 4 * NC;

  const int TB = 256;

  // symmetric normalization: deg (seeded 1.0 for self-loop) -> rsqrt
  k_fill <<<(N + TB - 1) / TB, TB, 0, stream>>>(dis, 1.0f, N);
  k_deg  <<<(E + TB - 1) / TB, TB, 0, stream>>>(dst, dis, E);
  k_rsqrt<<<(N + TB - 1) / TB, TB, 0, stream>>>(dis, N);

  const float* X = x;
  int K = F_IN;
  int nc = N * C_CH;
  int waves = (N + 15) / 16;
  int gthreads = waves * 32;
  for (int l = 0; l < 4; ++l) {
    if (K == 128)
      k_gemm_wmma<128><<<(gthreads + TB - 1) / TB, TB, 0, stream>>>(X, Wm[l], h, N);
    else
      k_gemm_wmma<64> <<<(gthreads + TB - 1) / TB, TB, 0, stream>>>(X, Wm[l], h, N);
    k_self<<<(nc + TB - 1) / TB, TB, 0, stream>>>(h, dis, bv[l], agg, N);
    int et = E * 16;
    k_edge<<<(et + TB - 1) / TB, TB, 0, stream>>>(src, dst, dis, h, agg, E);
    k_relu<<<(nc + TB - 1) / TB, TB, 0, stream>>>(agg, outs[l], nc);
    X = outs[l];
    K = C_CH;
  }

  k_fill<<<(G * C_CH + TB - 1) / TB, TB, 0, stream>>>(pp, 0.0f, G * C_CH);
  k_pool<<<(nc + TB - 1) / TB, TB, 0, stream>>>(outs[3], bat, pp, N);
  k_fc  <<<(G * CLS + TB - 1) / TB, TB, 0, stream>>>(pp, Wl, bl, zout, G, CLS);
}